// Carf_70720931496403
// MI455X (gfx1250) — compile-verified
//
#include <hip/hip_runtime.h>

// CA-CFAR 2D, single-pass tiled:
//   stage 1: global -> LDS tile load (62x94 region, zero-padded at borders)
//   stage 2: horizontal sliding-window sums (31-tap and 11-tap) in LDS
//   stage 3: vertical banded reduction via V_WMMA_F32_16X16X4_F32 (exact f32)
//   stage 4: combine + store both output channels
//
// Constants from reference: GUARD=5, BG=10 -> CFAR=31, INNER=11, HALF=15,
// AREA_FRONT=121, AREA_BACK=840.

#define IMG_H 1024
#define IMG_W 1024
#define TILE_Y 32
#define TILE_X 64
#define RIN_Y 62            // TILE_Y + 30
#define RIN_X 94            // TILE_X + 30
#define LIN_STRIDE 96       // padded row stride for input tile
#define HS_ROWS 64          // RIN_Y padded to multiple of 4-chunk reach (r0=16 + 47)
#define HS_STRIDE 72        // 2*72 % 64 == 16 -> paired-row B loads hit disjoint banks

typedef __attribute__((ext_vector_type(2))) float v2f;
typedef __attribute__((ext_vector_type(8))) float v8f;

__global__ __launch_bounds__(256)
void cfar_tile_kernel(const float* __restrict__ x, float* __restrict__ out,
                      int nbatch) {
    __shared__ float lin[RIN_Y * LIN_STRIDE];
    __shared__ float h31[HS_ROWS * HS_STRIDE];
    __shared__ float h11[HS_ROWS * HS_STRIDE];

    const int tid = threadIdx.x;
    const int tileX0 = blockIdx.x * TILE_X;
    const int tileY0 = blockIdx.y * TILE_Y;
    const int img = blockIdx.z;
    const float* __restrict__ xin = x + (size_t)img * IMG_H * IMG_W;

    // ---- Stage 1: load input region with zero padding -------------------
    for (int f = tid; f < RIN_Y * LIN_STRIDE; f += 256) {
        const int i = f / LIN_STRIDE;
        const int j = f - i * LIN_STRIDE;
        const int gy = tileY0 - 15 + i;
        const int gx = tileX0 - 15 + j;
        float v = 0.0f;
        if (j < RIN_X && gy >= 0 && gy < IMG_H && gx >= 0 && gx < IMG_W)
            v = xin[(size_t)gy * IMG_W + gx];
        lin[f] = v;
    }
    // zero the pad rows of the h-sum arrays (rows 62,63) so the zero-weight
    // band entries never multiply uninitialized LDS (NaN * 0 = NaN).
    for (int f = tid; f < 2 * HS_STRIDE; f += 256) {
        h31[RIN_Y * HS_STRIDE + f] = 0.0f;
        h11[RIN_Y * HS_STRIDE + f] = 0.0f;
    }
    __syncthreads();

    // ---- Stage 2: horizontal sliding sums (62 rows x 4 segments of 16) --
    if (tid < RIN_Y * 4) {
        const int row = tid >> 2;
        const int cseg = (tid & 3) * 16;
        const float* __restrict__ r = &lin[row * LIN_STRIDE];
        float s31 = 0.0f, s11 = 0.0f;
#pragma unroll
        for (int j = 0; j < 31; ++j) s31 += r[cseg + j];
#pragma unroll
        for (int j = 10; j <= 20; ++j) s11 += r[cseg + j];
        float* __restrict__ o31 = &h31[row * HS_STRIDE];
        float* __restrict__ o11 = &h11[row * HS_STRIDE];
#pragma unroll
        for (int c = 0; c < 16; ++c) {
            o31[cseg + c] = s31;
            o11[cseg + c] = s11;
            s31 += r[cseg + c + 31] - r[cseg + c];
            s11 += r[cseg + c + 21] - r[cseg + c + 10];
        }
    }
    __syncthreads();

    // ---- Stage 3: vertical banded reduction via WMMA f32 16x16x4 --------
    // Wave w owns 16x16 output subtile (sr = w/4, sc = w%4).
    // A layout (16x4 f32): lane L -> M = L%16, K = 2*(L/16)+v  (v = VGPR 0/1)
    // B layout (4x16 f32): lane L -> N = L%16, K = 2*(L/16)+v
    const int wave = tid >> 5;
    const int lane = tid & 31;
    const int sr = wave >> 2;            // 0..1
    const int sc = wave & 3;             // 0..3
    const int r0 = sr * 16;
    const int c0 = sc * 16;
    const int m = lane & 15;             // M for A, N for B
    const int khalf = (lane >> 4) * 2;   // 0 or 2
    const int bcol = c0 + m;

    v8f ctot = {};
    v8f cfr = {};

    // 31-tap: window for output row m covers h-rows r0+m .. r0+m+30
    // (k_g in [m, m+30]); K spans k_g = 0..45, padded to 48 -> 12 chunks.
#pragma unroll
    for (int q = 0; q < 12; ++q) {
        v2f a, b;
#pragma unroll
        for (int v = 0; v < 2; ++v) {
            const int kg = 4 * q + khalf + v;
            a[v] = (kg >= m && kg <= m + 30) ? 1.0f : 0.0f;
            b[v] = h31[(r0 + kg) * HS_STRIDE + bcol];
        }
        ctot = __builtin_amdgcn_wmma_f32_16x16x4_f32(
            false, a, false, b, (short)0, ctot, false, false);
    }

    // 11-tap: k_g in [m+10, m+20]; nonzero band only in chunks q = 2..8.
#pragma unroll
    for (int q = 2; q < 9; ++q) {
        v2f a, b;
#pragma unroll
        for (int v = 0; v < 2; ++v) {
            const int kg = 4 * q + khalf + v;
            a[v] = (kg >= m + 10 && kg <= m + 20) ? 1.0f : 0.0f;
            b[v] = h11[(r0 + kg) * HS_STRIDE + bcol];
        }
        cfr = __builtin_amdgcn_wmma_f32_16x16x4_f32(
            false, a, false, b, (short)0, cfr, false, false);
    }

    // ---- Stage 4: combine + store ---------------------------------------
    // D layout: VGPR j -> M = j + 8*(lane/16), N = lane%16
    const int mbase = (lane >> 4) * 8;
    const size_t base0 = (size_t)img * IMG_H * IMG_W;
    const size_t base1 = base0 + (size_t)nbatch * IMG_H * IMG_W;
#pragma unroll
    for (int j = 0; j < 8; ++j) {
        const int yy = tileY0 + r0 + mbase + j;
        const int xx = tileX0 + c0 + m;
        const float front = cfr[j] * (1.0f / 121.0f);
        const float backv = (ctot[j] - cfr[j]) * (1.0f / 840.0f);
        const size_t off = (size_t)yy * IMG_W + xx;
        out[base0 + off] = front / backv;   // channel 0: normalized ratio
        out[base1 + off] = front;           // channel 1: front / AREA_FRONT
    }
}

extern "C" void kernel_launch(void* const* d_in, const int* in_sizes, int n_in,
                              void* d_out, int out_size, void* d_ws, size_t ws_size,
                              hipStream_t stream) {
    (void)n_in; (void)out_size; (void)d_ws; (void)ws_size;
    const float* x = (const float*)d_in[0];
    float* out = (float*)d_out;
    const int nbatch = in_sizes[0] / (IMG_H * IMG_W);   // 32

    dim3 grid(IMG_W / TILE_X, IMG_H / TILE_Y, nbatch);  // 16 x 32 x 32
    dim3 block(256);
    cfar_tile_kernel<<<grid, block, 0, stream>>>(x, out, nbatch);
}